// HeteroGraphSAGE_57793079935345
// MI455X (gfx1250) — compile-verified
//
#include <hip/hip_runtime.h>
#include <cstdint>
#include <cstddef>

typedef float v2f __attribute__((ext_vector_type(2)));
typedef float v8f __attribute__((ext_vector_type(8)));

// ---------------------------------------------------------------------------
// WMMA GEMM:  Y[n x 64] = X[n x K] @ W[64 x K]^T  (+ broadcast bias[64], opt)
// One wave -> 16-row stripe x all 64 columns (4 accumulator tiles).
// Uses V_WMMA_F32_16X16X4_F32 (full fp32, matches fp32 reference).
// A frag (16x4 f32, ISA 7.12.2): lanes 0-15: v0=K0,v1=K1 ; lanes 16-31: K2,K3
// B frag (4x16): lanes 0-15 hold K0/K1 (cols N=lane), lanes 16-31 hold K2/K3
// C/D (16x16 f32): vgpr v <-> M = v (lanes 0-15) or 8+v (lanes 16-31), N=lane%16
// ---------------------------------------------------------------------------
template<int K>
__global__ __launch_bounds__(128)
void gemm_xwt_kernel(const float* __restrict__ X, const float* __restrict__ W,
                     const float* __restrict__ bias, float* __restrict__ Y, int n)
{
    constexpr int PK = K + 4;                 // LDS row pitch (avoid bank conflicts)
    __shared__ float sX[64 * PK];
    const int tid     = threadIdx.x;
    const int rowBase = blockIdx.x * 64;

    // cooperative, coalesced staging of a 64-row stripe of X (zero-padded)
    for (int i = tid; i < 64 * K; i += 128) {
        int r = i / K, c = i - r * K;
        int gr = rowBase + r;
        sX[r * PK + c] = (gr < n) ? X[(size_t)gr * K + c] : 0.0f;
    }
    __syncthreads();

    const int wave = tid >> 5;                // 0..3 -> which 16-row stripe
    const int lane = tid & 31;
    const int lr   = lane & 15;
    const int kOff = (lane < 16) ? 0 : 2;
    const int m0   = wave * 16;

    v8f acc[4];
#pragma unroll
    for (int t = 0; t < 4; ++t) {
        float bv = bias ? bias[t * 16 + lr] : 0.0f;   // column-broadcast bias
#pragma unroll
        for (int v = 0; v < 8; ++v) acc[t][v] = bv;
    }

#pragma unroll 4
    for (int k = 0; k < K; k += 4) {
        v2f a;
        a.x = sX[(m0 + lr) * PK + k + kOff];
        a.y = sX[(m0 + lr) * PK + k + kOff + 1];
#pragma unroll
        for (int t = 0; t < 4; ++t) {
            const float* wr = W + (size_t)(t * 16 + lr) * K + k + kOff;
            v2f b; b.x = wr[0]; b.y = wr[1];
            acc[t] = __builtin_amdgcn_wmma_f32_16x16x4_f32(
                false, a, false, b, (short)0, acc[t], false, false);
        }
    }

    const int mB = (lane < 16) ? 0 : 8;
    if (rowBase + 64 <= n) {
        // fast path: whole stripe in-bounds -> straight-line, unguarded stores
        float* yb = Y + (size_t)(rowBase + m0 + mB) * 64 + lr;
#pragma unroll
        for (int t = 0; t < 4; ++t) {
#pragma unroll
            for (int v = 0; v < 8; ++v)
                yb[(size_t)v * 64 + t * 16] = acc[t][v];
        }
    } else {
        // tail block only: per-element guard
#pragma unroll
        for (int t = 0; t < 4; ++t) {
            const int col = t * 16 + lr;
#pragma unroll
            for (int v = 0; v < 8; ++v) {
                int gr = rowBase + m0 + mB + v;
                if (gr < n) Y[(size_t)gr * 64 + col] = acc[t][v];
            }
        }
    }
}

// Wsum = sum of up to 3 relation slices of Wr[rel][64][K]; bsum likewise.
__global__ void sum_wr_kernel(const float* __restrict__ Wr, const float* __restrict__ b,
                              int K, int r0, int r1, int r2,
                              float* __restrict__ Wsum, float* __restrict__ bsum)
{
    int i  = blockIdx.x * blockDim.x + threadIdx.x;
    int sz = 64 * K;
    if (i < sz) {
        float s = Wr[(size_t)r0 * sz + i];
        if (r1 >= 0) s += Wr[(size_t)r1 * sz + i];
        if (r2 >= 0) s += Wr[(size_t)r2 * sz + i];
        Wsum[i] = s;
    }
    if (i < 64) {
        float s = b[r0 * 64 + i];
        if (r1 >= 0) s += b[r1 * 64 + i];
        if (r2 >= 0) s += b[r2 * 64 + i];
        bsum[i] = s;
    }
}

// One thread per (edge, channel-pair): agg[dst][2c..2c+1] += tmp[src][2c..2c+1].
// Atomic f32 adds are mandatory (overlapping dst); float2 loads halve the
// per-thread index arithmetic on the bandwidth-dominant kernel.
__global__ void scatter_kernel(const int* __restrict__ src, const int* __restrict__ dst,
                               const float* __restrict__ t, float* __restrict__ agg,
                               float* __restrict__ cnt, long long nE)
{
    long long i = (long long)blockIdx.x * blockDim.x + threadIdx.x;
    if (i >= nE * 32) return;
    int e = (int)(i >> 5), c = (int)(i & 31);      // 32 float2-slots per edge
    int s = src[e], d = dst[e];
    float2 v = *(const float2*)(t + (size_t)s * 64 + c * 2);
    float*  ap = agg + (size_t)d * 64 + c * 2;
    atomicAdd(ap,     v.x);
    atomicAdd(ap + 1, v.y);
    if (c == 0) atomicAdd(&cnt[d], 1.0f);
}

// acc += agg / max(cnt,1)
__global__ void add_mean_kernel(const float* __restrict__ agg, const float* __restrict__ cnt,
                                float* __restrict__ acc, long long n64)
{
    long long i = (long long)blockIdx.x * blockDim.x + threadIdx.x;
    if (i >= n64) return;
    float k = cnt[i >> 6];
    k = (k > 1.0f) ? k : 1.0f;
    acc[i] += agg[i] / k;
}

// a = (relu?)(a * s)   -- relation-mean scale and optional ReLU
__global__ void finish_kernel(float* __restrict__ a, long long n, float s, int relu)
{
    long long i = (long long)blockIdx.x * blockDim.x + threadIdx.x;
    if (i >= n) return;
    float v = a[i] * s;
    if (relu) v = fmaxf(v, 0.0f);
    a[i] = v;
}

// row-wise L2 normalize (rows of 64 floats); one wave32 per row (2 floats/lane)
__global__ __launch_bounds__(256)
void l2norm_kernel(float* __restrict__ x, int n)
{
    int row  = blockIdx.x * 8 + (threadIdx.x >> 5);
    int lane = threadIdx.x & 31;
    if (row >= n) return;
    float2* p = (float2*)(x + (size_t)row * 64);
    float2 v = p[lane];
    float ss = v.x * v.x + v.y * v.y;
#pragma unroll
    for (int o = 16; o; o >>= 1) ss += __shfl_xor(ss, o, 32);
    float inv = 1.0f / fmaxf(sqrtf(ss), 1e-12f);
    v.x *= inv; v.y *= inv;
    p[lane] = v;
}

extern "C" void kernel_launch(void* const* d_in, const int* in_sizes, int n_in,
                              void* d_out, int out_size, void* d_ws, size_t ws_size,
                              hipStream_t stream)
{
    (void)n_in; (void)out_size; (void)ws_size;

    const float* xin[3] = { (const float*)d_in[0], (const float*)d_in[1], (const float*)d_in[2] };
    int nNode[3] = { in_sizes[0] / 128, in_sizes[1] / 128, in_sizes[2] / 128 };

    // relation table: idx -> (src type, dst type, edge input slot)
    const int relSrc[8] = {0,1,0,1,2,2,0,1};
    const int relDst[8] = {1,2,2,0,1,0,0,1};
    const int relEin[8] = {3,4,5,6,7,8,9,10};

    const float* W1l = (const float*)d_in[11];
    const float* W1r = (const float*)d_in[12];
    const float* b1  = (const float*)d_in[13];
    const float* W2l = (const float*)d_in[14];
    const float* W2r = (const float*)d_in[15];
    const float* b2  = (const float*)d_in[16];
    const float* W3l = (const float*)d_in[17];
    const float* W3r = (const float*)d_in[18];
    const float* b3  = (const float*)d_in[19];

    // ---- workspace carve (floats) ----
    float* p = (float*)d_ws;
    auto carve = [&](size_t nf) { float* q = p; p += nf; return q; };
    float* hA[3]; float* hB[3];
    for (int i = 0; i < 3; ++i) hA[i] = carve((size_t)nNode[i] * 64);
    for (int i = 0; i < 3; ++i) hB[i] = carve((size_t)nNode[i] * 64);
    float* tmpS = carve((size_t)nNode[0] * 64);   // transformed source feats
    float* tmpA = carve((size_t)nNode[0] * 64);   // per-relation agg sums
    float* cnt  = carve((size_t)nNode[0]);        // per-relation in-degree
    float* Wsum = carve(64 * 128);
    float* bsum = carve(64);

    float* outP[3] = { (float*)d_out,
                       (float*)d_out + (size_t)nNode[0] * 64,
                       (float*)d_out + (size_t)(nNode[0] + nNode[1]) * 64 };

    auto launch_gemm = [&](int K, const float* X, const float* W, const float* bias,
                           float* Y, int n) {
        int blocks = (n + 63) / 64;
        if (K == 128) gemm_xwt_kernel<128><<<blocks, 128, 0, stream>>>(X, W, bias, Y, n);
        else          gemm_xwt_kernel<64> <<<blocks, 128, 0, stream>>>(X, W, bias, Y, n);
    };

    auto run_layer = [&](const float* const xs[3], float* const acc[3], int K,
                         const float* Wl, const float* Wr, const float* b,
                         int nRel, int lastLayer) {
        // 1) dst-side: acc_D = x_D @ (sum_r Wr[r])^T + sum_r b[r]
        int intoC[3] = {3, 5, (nRel == 8) ? 6 : -1};
        int intoM[3] = {0, 4, (nRel == 8) ? 7 : -1};
        int intoD[3] = {1, 2, -1};
        const int* lists[3] = { intoC, intoM, intoD };
        for (int dt = 0; dt < 3; ++dt) {
            const int* L = lists[dt];
            int sb = (64 * K + 255) / 256;
            sum_wr_kernel<<<sb, 256, 0, stream>>>(Wr, b, K, L[0], L[1], L[2], Wsum, bsum);
            launch_gemm(K, xs[dt], Wsum, bsum, acc[dt], nNode[dt]);
        }
        // 2) per-relation: transform sources, segment-mean, accumulate
        for (int r = 0; r < nRel; ++r) {
            int S = relSrc[r], D = relDst[r];
            const int* eb = (const int*)d_in[relEin[r]];
            long long E = in_sizes[relEin[r]] / 2;
            launch_gemm(K, xs[S], Wl + (size_t)r * 64 * K, nullptr, tmpS, nNode[S]);
            hipMemsetAsync(tmpA, 0, (size_t)nNode[D] * 64 * sizeof(float), stream);
            hipMemsetAsync(cnt,  0, (size_t)nNode[D] * sizeof(float), stream);
            long long tot = E * 32;
            scatter_kernel<<<(int)((tot + 255) / 256), 256, 0, stream>>>(
                eb, eb + E, tmpS, tmpA, cnt, E);
            long long n64 = (long long)nNode[D] * 64;
            add_mean_kernel<<<(int)((n64 + 255) / 256), 256, 0, stream>>>(
                tmpA, cnt, acc[D], n64);
        }
        // 3) relation-mean (+ ReLU on hidden layers)
        float Rv[3] = { (nRel == 8) ? 3.0f : 2.0f, (nRel == 8) ? 3.0f : 2.0f, 2.0f };
        for (int dt = 0; dt < 3; ++dt) {
            long long n64 = (long long)nNode[dt] * 64;
            finish_kernel<<<(int)((n64 + 255) / 256), 256, 0, stream>>>(
                acc[dt], n64, 1.0f / Rv[dt], lastLayer ? 0 : 1);
        }
    };

    // Layer 1: IN=128 -> 64, inputs from d_in, acc in hA
    { const float* xs[3] = { xin[0], xin[1], xin[2] };
      float* ac[3] = { hA[0], hA[1], hA[2] };
      run_layer(xs, ac, 128, W1l, W1r, b1, 6, 0); }
    // Layer 2: 64 -> 64, hA -> hB
    { const float* xs[3] = { hA[0], hA[1], hA[2] };
      float* ac[3] = { hB[0], hB[1], hB[2] };
      run_layer(xs, ac, 64, W2l, W2r, b2, 6, 0); }
    // Layer 3: 64 -> 64 (8 relations), hB -> d_out regions
    { const float* xs[3] = { hB[0], hB[1], hB[2] };
      float* ac[3] = { outP[0], outP[1], outP[2] };
      run_layer(xs, ac, 64, W3l, W3r, b3, 8, 1); }

    // Final row-wise L2 normalize, in place on d_out
    for (int dt = 0; dt < 3; ++dt) {
        int blocks = (nNode[dt] + 7) / 8;
        l2norm_kernel<<<blocks, 256, 0, stream>>>(outP[dt], nNode[dt]);
    }
}